// FusionIPOT_49435073577567
// MI455X (gfx1250) — compile-verified
//
#include <hip/hip_runtime.h>

// ---------------- problem constants ----------------
#define BSZ     2
#define NP      2048
#define CH      256
#define FFN_CH  1024
#define NCLS_   10
#define ITERS   100
#define NCHUNK  128            // column-pass row chunks (deterministic 2-stage reduce)
#define RCH     (NP / NCHUNK)  // 16 rows per chunk

typedef float v2f __attribute__((ext_vector_type(2)));
typedef float v8f __attribute__((ext_vector_type(8)));

// ---------------- workspace layout (float offsets) ----------------
static const size_t OFF_Q     = 0;                       // BSZ*NP*NP = 8388608
static const size_t OFF_G     = 8388608;                 // BSZ*NP*NP (dead after IPOT loop)
static const size_t OFF_H     = OFF_G;                   // alias: BSZ*NP*FFN = 4194304
static const size_t OFF_Y     = OFF_G + 4194304;         // BSZ*NP*CH = 1048576
static const size_t OFF_Z     = OFF_G + 5242880;         // BSZ*NP*CH
static const size_t OFF_PF    = 16777216;                // BSZ*CH*NP
static const size_t OFF_IMF   = OFF_PF   + 1048576;
static const size_t OFF_IMFT  = OFF_IMF  + 1048576;      // BSZ*NP*CH
static const size_t OFF_X     = OFF_IMFT + 1048576;      // BSZ*NP*2CH = 2097152
static const size_t OFF_X1    = OFF_X    + 2097152;      // BSZ*NP*CH
static const size_t OFF_PART  = OFF_X1   + 1048576;      // BSZ*NCHUNK*NP = 524288
static const size_t OFF_PCLS  = OFF_PART + 524288;       // BSZ*NP
static const size_t OFF_ICLS  = OFF_PCLS + 4096;
static const size_t OFF_A     = OFF_ICLS + 4096;
static const size_t OFF_B     = OFF_A    + 4096;
static const size_t OFF_SCALE = OFF_B    + 4096;
static const size_t OFF_WOUTT = OFF_SCALE+ 4096;         // 512*256
static const size_t OFF_WF1T  = OFF_WOUTT+ 131072;       // 256*1024
static const size_t OFF_WF2T  = OFF_WF1T + 262144;       // 1024*256

// ================== small elementwise kernels ==================
__global__ __launch_bounds__(256) void k_sigmax(const float* __restrict__ cls,
                                                float* __restrict__ out) {
  int idx = blockIdx.x * 256 + threadIdx.x;   // BSZ*NP
  int b = idx >> 11, n = idx & (NP - 1);
  float m = -1e30f;
  for (int k = 0; k < NCLS_; ++k)
    m = fmaxf(m, cls[((size_t)b * NCLS_ + k) * NP + n]);
  out[idx] = 1.0f / (1.0f + expf(-m));        // max(sigmoid) == sigmoid(max)
}

__global__ __launch_bounds__(256) void k_init_vecs(float* __restrict__ avec,
                                                   float* __restrict__ bvec) {
  int idx = blockIdx.x * 256 + threadIdx.x;
  avec[idx] = 1.0f;
  bvec[idx] = 1.0f / (float)NP;               // b0 = 1/N
}

__global__ __launch_bounds__(256) void k_init_G(const float* __restrict__ ppos,
                                                const float* __restrict__ ipos,
                                                float* __restrict__ G,
                                                float* __restrict__ Q) {
  size_t idx = (size_t)blockIdx.x * 256 + threadIdx.x;   // BSZ*NP*NP/4
  const size_t per_b = (size_t)NP * NP / 4;
  int b = (int)(idx / per_b);
  size_t r = idx % per_b;
  int n  = (int)(r / (NP / 4));
  int m4 = (int)(r % (NP / 4));
  float px = ppos[((size_t)b * NP + n) * 2 + 0];
  float py = ppos[((size_t)b * NP + n) * 2 + 1];
  float4 g, q;
  float* gp = (float*)&g; float* qp = (float*)&q;
#pragma unroll
  for (int j = 0; j < 4; ++j) {
    int m = m4 * 4 + j;
    float dx = px - ipos[((size_t)b * NP + m) * 2 + 0];
    float dy = py - ipos[((size_t)b * NP + m) * 2 + 1];
    float gg = expf(-sqrtf(dx * dx + dy * dy) * 0.1f);
    gp[j] = gg;
    qp[j] = fmaxf(gg, 1e-6f);                 // Q_1 = max(G*res0, 1e-6), res0 = 1
  }
  ((float4*)G)[idx] = g;
  ((float4*)Q)[idx] = q;
}

// ================== IPOT loop kernels (L2-resident streaming) ==================
// Row pass: (optionally) fold Q <- max(G * a_prev * Q * b, 1e-6), then a = pcls/(Q@b + eps)
__global__ __launch_bounds__(256) void k_row_pass(const float* __restrict__ G,
                                                  float* __restrict__ Q,
                                                  const float* __restrict__ bvec,
                                                  float* __restrict__ avec,
                                                  const float* __restrict__ pcls,
                                                  int fold) {
  int row = blockIdx.x;                        // BSZ*NP
  int b = row >> 11, tid = threadIdx.x;
  const float4* g4 = (const float4*)(G + (size_t)row * NP);
  float4*       q4 = (float4*)(Q + (size_t)row * NP);
  const float4* b4 = (const float4*)(bvec + (size_t)b * NP);
  float aprev = avec[row];
  float acc = 0.0f;
  for (int i = tid; i < NP / 4; i += 256) {
    float4 q = q4[i];
    float4 bb = b4[i];
    if (fold) {
      float4 g = g4[i];
      float s = aprev;
      q.x = fmaxf(g.x * s * q.x * bb.x, 1e-6f);
      q.y = fmaxf(g.y * s * q.y * bb.y, 1e-6f);
      q.z = fmaxf(g.z * s * q.z * bb.z, 1e-6f);
      q.w = fmaxf(g.w * s * q.w * bb.w, 1e-6f);
      q4[i] = q;
    }
    acc += q.x * bb.x + q.y * bb.y + q.z * bb.z + q.w * bb.w;
  }
  __shared__ float red[256];
  red[tid] = acc; __syncthreads();
  for (int s = 128; s > 0; s >>= 1) { if (tid < s) red[tid] += red[tid + s]; __syncthreads(); }
  if (tid == 0) avec[row] = pcls[row] / (red[0] + 1e-6f);
}

// Column pass stage 1: deterministic per-chunk partial column sums of Q^T a
__global__ __launch_bounds__(256) void k_col_partial(const float* __restrict__ Q,
                                                     const float* __restrict__ avec,
                                                     float* __restrict__ partial) {
  __shared__ __align__(16) float lsum[NP];
  int bx = blockIdx.x;                         // BSZ*NCHUNK
  int b = bx / NCHUNK, ch = bx % NCHUNK;
  int tid = threadIdx.x;
  for (int j = tid; j < NP / 4; j += 256) ((float4*)lsum)[j] = make_float4(0, 0, 0, 0);
  int r0 = ch * RCH;
  for (int n = r0; n < r0 + RCH; ++n) {
    float an = avec[(size_t)b * NP + n];
    const float4* qrow = (const float4*)(Q + ((size_t)b * NP + n) * NP);
    for (int j = tid; j < NP / 4; j += 256) {  // strict per-thread LDS ownership: no barriers
      float4 q = qrow[j];
      float4 s = ((float4*)lsum)[j];
      s.x += q.x * an; s.y += q.y * an; s.z += q.z * an; s.w += q.w * an;
      ((float4*)lsum)[j] = s;
    }
  }
  float4* pp = (float4*)(partial + (size_t)bx * NP);
  for (int j = tid; j < NP / 4; j += 256) pp[j] = ((float4*)lsum)[j];
}

// Column pass stage 2: fixed-order reduce -> b = icls/(colsum + eps)
__global__ __launch_bounds__(256) void k_col_finalize(const float* __restrict__ partial,
                                                      const float* __restrict__ icls,
                                                      float* __restrict__ bvec) {
  int idx = blockIdx.x * 256 + threadIdx.x;    // BSZ*NP
  int b = idx >> 11, m = idx & (NP - 1);
  float s = 0.0f;
  for (int ch = 0; ch < NCHUNK; ++ch)
    s += partial[((size_t)(b * NCHUNK + ch)) * NP + m];
  bvec[idx] = icls[idx] / (s + 1e-6f);
}

// Final row scale: res_norm[n,m] = scale[n]*Q[n,m]*b[m], scale = a/max(a*Σ Qb, 1e-12)
__global__ __launch_bounds__(256) void k_row_final(const float* __restrict__ Q,
                                                   const float* __restrict__ bvec,
                                                   const float* __restrict__ avec,
                                                   float* __restrict__ scale) {
  int row = blockIdx.x;
  int b = row >> 11, tid = threadIdx.x;
  const float4* q4 = (const float4*)(Q + (size_t)row * NP);
  const float4* b4 = (const float4*)(bvec + (size_t)b * NP);
  float acc = 0.0f;
  for (int i = tid; i < NP / 4; i += 256) {
    float4 q = q4[i], bb = b4[i];
    acc += q.x * bb.x + q.y * bb.y + q.z * bb.z + q.w * bb.w;
  }
  __shared__ float red[256];
  red[tid] = acc; __syncthreads();
  for (int s = 128; s > 0; s >>= 1) { if (tid < s) red[tid] += red[tid + s]; __syncthreads(); }
  if (tid == 0) {
    float a = avec[row];
    scale[row] = a / fmaxf(a * red[0], 1e-12f);
  }
}

// ================== WMMA fp32 GEMM (V_WMMA_F32_16X16X4_F32) ==================
// C[M,Nc] = A[M,K] x B[K,Nc]  (row-major), batched via strides.
// Options: bvec (scale A along K), rowscale, bias (per col), resid, relu.
// WG = 256 thr = 8 waves (2x4), tile 64x128, K-chunk 32 through padded LDS.
#define AS_LD 36
#define BS_LD 136
__global__ __launch_bounds__(256) void k_wmma_gemm(
    const float* __restrict__ A, const float* __restrict__ B, float* __restrict__ C,
    int M, int Nc, int K, int lda, int ldb, int ldc,
    long long strideA, long long strideB, long long strideC,
    const float* __restrict__ bvec, long long strideBv,
    const float* __restrict__ rowscale, long long strideRs,
    const float* __restrict__ bias,
    const float* __restrict__ resid, long long strideRes, int ldres,
    int relu) {
  __shared__ __align__(16) float As[64 * AS_LD];
  __shared__ __align__(16) float Bs[32 * BS_LD];
  int bz = blockIdx.z;
  A += (size_t)bz * strideA;
  B += (size_t)bz * strideB;
  C += (size_t)bz * strideC;
  const float* bv  = bvec     ? bvec     + (size_t)bz * strideBv  : nullptr;
  const float* rs  = rowscale ? rowscale + (size_t)bz * strideRs  : nullptr;
  const float* rsd = resid    ? resid    + (size_t)bz * strideRes : nullptr;

  int tid = threadIdx.x;
  int lane = tid & 31;
  int wave = tid >> 5;
  int hf = lane >> 4, lr = lane & 15;
  int wm = wave >> 2, wn = wave & 3;             // 2 x 4 wave grid
  int rowBase = blockIdx.y * 64;
  int colBase = blockIdx.x * 128;

  int ar = tid >> 2, aq = tid & 3;               // A staging: 4 thr/row, 2 float4 each
  int br = tid >> 3, bj = tid & 7;               // B staging: 8 thr/row, 4 float4 each

  v8f acc[2][2] = {};

  for (int k0 = 0; k0 < K; k0 += 32) {
    __syncthreads();
    // ---- stage A tile 64x32 ----
    {
      const float* ap = A + (size_t)(rowBase + ar) * lda + k0 + aq * 4;
      float4 a0 = *(const float4*)(ap);
      float4 a1 = *(const float4*)(ap + 16);
      if (bv) {
        float4 b0 = *(const float4*)(bv + k0 + aq * 4);
        float4 b1 = *(const float4*)(bv + k0 + aq * 4 + 16);
        a0.x *= b0.x; a0.y *= b0.y; a0.z *= b0.z; a0.w *= b0.w;
        a1.x *= b1.x; a1.y *= b1.y; a1.z *= b1.z; a1.w *= b1.w;
      }
      *(float4*)(&As[ar * AS_LD + aq * 4])      = a0;
      *(float4*)(&As[ar * AS_LD + aq * 4 + 16]) = a1;
    }
    // ---- stage B tile 32x128 ----
#pragma unroll
    for (int u = 0; u < 4; ++u) {
      int cc = 4 * (bj + 8 * u);
      float4 v = *(const float4*)(B + (size_t)(k0 + br) * ldb + colBase + cc);
      *(float4*)(&Bs[br * BS_LD + cc]) = v;
    }
    __syncthreads();
    // ---- 8 micro-steps of K=4 ----
#pragma unroll
    for (int kk = 0; kk < 32; kk += 4) {
      v2f af[2], bfr[2];
#pragma unroll
      for (int mi = 0; mi < 2; ++mi) {
        int r = wm * 32 + mi * 16 + lr;          // A frag: lane 0-15 K{0,1}, 16-31 K{2,3}
        af[mi].x = As[r * AS_LD + kk + hf * 2];
        af[mi].y = As[r * AS_LD + kk + hf * 2 + 1];
      }
#pragma unroll
      for (int ni = 0; ni < 2; ++ni) {
        int c = wn * 32 + ni * 16 + lr;          // B frag: mirrored layout
        bfr[ni].x = Bs[(kk + hf * 2) * BS_LD + c];
        bfr[ni].y = Bs[(kk + hf * 2 + 1) * BS_LD + c];
      }
#pragma unroll
      for (int mi = 0; mi < 2; ++mi)
#pragma unroll
        for (int ni = 0; ni < 2; ++ni)
          acc[mi][ni] = __builtin_amdgcn_wmma_f32_16x16x4_f32(
              false, af[mi], false, bfr[ni], (short)0, acc[mi][ni], false, false);
    }
  }
  // ---- epilogue (C/D layout: lane = col, VGPR r = row r + 8*half) ----
#pragma unroll
  for (int mi = 0; mi < 2; ++mi)
#pragma unroll
    for (int ni = 0; ni < 2; ++ni) {
      int col = colBase + wn * 32 + ni * 16 + lr;
      float bcol = bias ? bias[col] : 0.0f;
#pragma unroll
      for (int r = 0; r < 8; ++r) {
        int row = rowBase + wm * 32 + mi * 16 + 8 * hf + r;
        float v = acc[mi][ni][r] + bcol;
        if (relu) v = fmaxf(v, 0.0f);
        if (rsd)  v += rsd[(size_t)row * ldres + col];
        if (rs)   v *= rs[row];
        C[(size_t)row * ldc + col] = v;
      }
    }
}

// ================== transposes / layernorms / head ==================
__global__ __launch_bounds__(256) void k_transpose(const float* __restrict__ in,
                                                   float* __restrict__ out, int R, int Cc) {
  __shared__ float tile[32][33];
  int c0 = blockIdx.x * 32, r0 = blockIdx.y * 32;
  int tx = threadIdx.x & 31, ty = threadIdx.x >> 5;
#pragma unroll
  for (int yy = 0; yy < 32; yy += 8)
    tile[ty + yy][tx] = in[(size_t)(r0 + ty + yy) * Cc + c0 + tx];
  __syncthreads();
#pragma unroll
  for (int yy = 0; yy < 32; yy += 8)
    out[(size_t)(c0 + ty + yy) * R + r0 + tx] = tile[tx][ty + yy];
}

// raw [C][N] + bias1 + bias2 + Wpos@pos^T  -> out [N][ldo] (transposed)
__global__ __launch_bounds__(256) void k_addpos_t(const float* __restrict__ raw,
                                                  const float* __restrict__ pos,
                                                  const float* __restrict__ Wpos,
                                                  const float* __restrict__ b1,
                                                  const float* __restrict__ b2,
                                                  float* __restrict__ out, int ldo,
                                                  long long strideOut) {
  __shared__ float tile[32][33];
  int n0 = blockIdx.x * 32, c0 = blockIdx.y * 32, b = blockIdx.z;
  const float* rawb = raw + (size_t)b * CH * NP;
  const float* posb = pos + (size_t)b * NP * 2;
  float* outb = out + (size_t)b * strideOut;
  int tx = threadIdx.x & 31, ty = threadIdx.x >> 5;
#pragma unroll
  for (int yy = 0; yy < 32; yy += 8) {
    int c = c0 + ty + yy, n = n0 + tx;
    float v = rawb[(size_t)c * NP + n] + b1[c] + b2[c] +
              Wpos[c * 2 + 0] * posb[n * 2 + 0] + Wpos[c * 2 + 1] * posb[n * 2 + 1];
    tile[ty + yy][tx] = v;
  }
  __syncthreads();
#pragma unroll
  for (int yy = 0; yy < 32; yy += 8) {
    int n = n0 + ty + yy, c = c0 + tx;
    outb[(size_t)n * ldo + c] = tile[tx][ty + yy];
  }
}

__global__ __launch_bounds__(256) void k_ln(const float* __restrict__ in,
                                            float* __restrict__ out,
                                            const float* __restrict__ g,
                                            const float* __restrict__ bta) {
  __shared__ float red[256];
  int row = blockIdx.x, tid = threadIdx.x;      // BSZ*NP rows, C=256 per row
  float v = in[(size_t)row * CH + tid];
  red[tid] = v; __syncthreads();
  for (int s = 128; s > 0; s >>= 1) { if (tid < s) red[tid] += red[tid + s]; __syncthreads(); }
  float mu = red[0] * (1.0f / CH);
  __syncthreads();
  float d = v - mu;
  red[tid] = d * d; __syncthreads();
  for (int s = 128; s > 0; s >>= 1) { if (tid < s) red[tid] += red[tid + s]; __syncthreads(); }
  float var = red[0] * (1.0f / CH);
  out[(size_t)row * CH + tid] = d * rsqrtf(var + 1e-5f) * g[tid] + bta[tid];
}

// final LN + transposed feature output + prediction head + center/new_pos
__global__ __launch_bounds__(256) void k_ln_final(const float* __restrict__ in,
                                                  const float* __restrict__ g,
                                                  const float* __restrict__ bta,
                                                  const float* __restrict__ Wpred,
                                                  const float* __restrict__ bpred,
                                                  const float* __restrict__ ppos,
                                                  float* __restrict__ out_x,
                                                  float* __restrict__ out_np,
                                                  float* __restrict__ out_center) {
  __shared__ float red[256];
  int row = blockIdx.x, tid = threadIdx.x;
  int b = row >> 11, n = row & (NP - 1);
  float v = in[(size_t)row * CH + tid];
  red[tid] = v; __syncthreads();
  for (int s = 128; s > 0; s >>= 1) { if (tid < s) red[tid] += red[tid + s]; __syncthreads(); }
  float mu = red[0] * (1.0f / CH);
  __syncthreads();
  float d = v - mu;
  red[tid] = d * d; __syncthreads();
  for (int s = 128; s > 0; s >>= 1) { if (tid < s) red[tid] += red[tid + s]; __syncthreads(); }
  float var = red[0] * (1.0f / CH);
  float xv = d * rsqrtf(var + 1e-5f) * g[tid] + bta[tid];
  out_x[((size_t)b * CH + tid) * NP + n] = xv;   // [bs, C, N]
  __syncthreads();
  red[tid] = Wpred[tid] * xv; __syncthreads();
  for (int s = 128; s > 0; s >>= 1) { if (tid < s) red[tid] += red[tid + s]; __syncthreads(); }
  float d0 = red[0];
  __syncthreads();
  red[tid] = Wpred[CH + tid] * xv; __syncthreads();
  for (int s = 128; s > 0; s >>= 1) { if (tid < s) red[tid] += red[tid + s]; __syncthreads(); }
  if (tid == 0) {
    float d1 = red[0];
    float c0 = d0 + bpred[0] + ppos[((size_t)b * NP + n) * 2 + 0];
    float c1 = d1 + bpred[1] + ppos[((size_t)b * NP + n) * 2 + 1];
    out_center[((size_t)b * 2 + 0) * NP + n] = c0;
    out_center[((size_t)b * 2 + 1) * NP + n] = c1;
    out_np[((size_t)b * NP + n) * 2 + 0] = c0;
    out_np[((size_t)b * NP + n) * 2 + 1] = c1;
  }
}

// ================== host launcher ==================
extern "C" void kernel_launch(void* const* d_in, const int* in_sizes, int n_in,
                              void* d_out, int out_size, void* d_ws, size_t ws_size,
                              hipStream_t stream) {
  const float* ptsf    = (const float*)d_in[0];
  const float* ppos    = (const float*)d_in[1];
  const float* imgf    = (const float*)d_in[2];
  const float* ipos    = (const float*)d_in[3];
  const float* pts_cls = (const float*)d_in[4];
  const float* img_cls = (const float*)d_in[5];
  const float* Wp   = (const float*)d_in[6];
  const float* bp   = (const float*)d_in[7];
  const float* Wi   = (const float*)d_in[8];
  const float* bi   = (const float*)d_in[9];
  const float* Wqp  = (const float*)d_in[10];
  const float* bqp  = (const float*)d_in[11];
  const float* Wkp  = (const float*)d_in[12];
  const float* bkp  = (const float*)d_in[13];
  const float* Wout = (const float*)d_in[14];
  const float* bout = (const float*)d_in[15];
  const float* g_out    = (const float*)d_in[16];
  const float* beta_out = (const float*)d_in[17];
  const float* Wf1  = (const float*)d_in[18];
  const float* bf1  = (const float*)d_in[19];
  const float* Wf2  = (const float*)d_in[20];
  const float* bf2  = (const float*)d_in[21];
  const float* g_ln = (const float*)d_in[22];
  const float* b_ln = (const float*)d_in[23];
  const float* Wpred = (const float*)d_in[24];
  const float* bpred = (const float*)d_in[25];

  float* ws = (float*)d_ws;
  float* Q      = ws + OFF_Q;
  float* G      = ws + OFF_G;
  float* H      = ws + OFF_H;
  float* Y      = ws + OFF_Y;
  float* Z      = ws + OFF_Z;
  float* pf_raw = ws + OFF_PF;
  float* imf_raw= ws + OFF_IMF;
  float* imfT   = ws + OFF_IMFT;
  float* X      = ws + OFF_X;
  float* X1     = ws + OFF_X1;
  float* part   = ws + OFF_PART;
  float* pcls   = ws + OFF_PCLS;
  float* icls   = ws + OFF_ICLS;
  float* avec   = ws + OFF_A;
  float* bvec   = ws + OFF_B;
  float* scl    = ws + OFF_SCALE;
  float* WoutT  = ws + OFF_WOUTT;
  float* Wf1T   = ws + OFF_WF1T;
  float* Wf2T   = ws + OFF_WF2T;

  float* out_x      = (float*)d_out;                 // [2,256,2048]
  float* out_np     = out_x + (size_t)BSZ * CH * NP; // [2,2048,2]
  float* out_center = out_np + (size_t)BSZ * NP * 2; // [2,2,2048]

  auto gemm = [&](const float* A, const float* B, float* C, int M, int Nc, int K,
                  int lda, int ldb, int ldc,
                  long long sA, long long sB, long long sC,
                  const float* bv, long long sBv,
                  const float* rsc, long long sRs,
                  const float* bias, const float* resid, long long sRes, int ldres,
                  int relu) {
    dim3 grid(Nc / 128, M / 64, BSZ);
    k_wmma_gemm<<<grid, dim3(256), 0, stream>>>(A, B, C, M, Nc, K, lda, ldb, ldc,
                                                sA, sB, sC, bv, sBv, rsc, sRs,
                                                bias, resid, sRes, ldres, relu);
  };

  // ---- setup ----
  k_sigmax<<<BSZ * NP / 256, 256, 0, stream>>>(pts_cls, pcls);
  k_sigmax<<<BSZ * NP / 256, 256, 0, stream>>>(img_cls, icls);
  k_init_vecs<<<BSZ * NP / 256, 256, 0, stream>>>(avec, bvec);
  k_init_G<<<(int)((size_t)BSZ * NP * NP / 4 / 256), 256, 0, stream>>>(ppos, ipos, G, Q);

  // projections pf = Wp@ptsf, imf = Wi@imgf  (weights shared across batch: strideA=0)
  gemm(Wp, ptsf, pf_raw, CH, NP, CH, CH, NP, NP,
       0, (long long)CH * NP, (long long)CH * NP,
       nullptr, 0, nullptr, 0, nullptr, nullptr, 0, 0, 0);
  gemm(Wi, imgf, imf_raw, CH, NP, CH, CH, NP, NP,
       0, (long long)CH * NP, (long long)CH * NP,
       nullptr, 0, nullptr, 0, nullptr, nullptr, 0, 0, 0);

  // + pos embedding + biases, transposed into X[:, :256] and imfT
  k_addpos_t<<<dim3(NP / 32, CH / 32, BSZ), 256, 0, stream>>>(
      pf_raw, ppos, Wqp, bp, bqp, X, 2 * CH, (long long)NP * 2 * CH);
  k_addpos_t<<<dim3(NP / 32, CH / 32, BSZ), 256, 0, stream>>>(
      imf_raw, ipos, Wkp, bi, bkp, imfT, CH, (long long)NP * CH);

  // weight transposes (row-major B for the WMMA GEMM)
  k_transpose<<<dim3(2 * CH / 32, CH / 32), 256, 0, stream>>>(Wout, WoutT, CH, 2 * CH);
  k_transpose<<<dim3(CH / 32, FFN_CH / 32), 256, 0, stream>>>(Wf1, Wf1T, FFN_CH, CH);
  k_transpose<<<dim3(FFN_CH / 32, CH / 32), 256, 0, stream>>>(Wf2, Wf2T, CH, FFN_CH);

  // ---- IPOT loop: Q/G stay L2-resident (64 MB < 192 MB L2) ----
  for (int t = 1; t <= ITERS; ++t) {
    k_row_pass<<<BSZ * NP, 256, 0, stream>>>(G, Q, bvec, avec, pcls, (t > 1) ? 1 : 0);
    k_col_partial<<<BSZ * NCHUNK, 256, 0, stream>>>(Q, avec, part);
    k_col_finalize<<<BSZ * NP / 256, 256, 0, stream>>>(part, icls, bvec);
  }
  k_row_final<<<BSZ * NP, 256, 0, stream>>>(Q, bvec, avec, scl);

  // ---- attention apply: X[:,256:512] = rowscale * (Q .* b) @ imfT ----
  gemm(Q, imfT, X + CH, NP, CH, NP, NP, CH, 2 * CH,
       (long long)NP * NP, (long long)NP * CH, (long long)NP * 2 * CH,
       bvec, NP, scl, NP, nullptr, nullptr, 0, 0, 0);

  // ---- out projection + LN ----
  gemm(X, WoutT, Y, NP, CH, 2 * CH, 2 * CH, CH, CH,
       (long long)NP * 2 * CH, 0, (long long)NP * CH,
       nullptr, 0, nullptr, 0, bout, nullptr, 0, 0, 0);
  k_ln<<<BSZ * NP, 256, 0, stream>>>(Y, X1, g_out, beta_out);

  // ---- FFN ----
  gemm(X1, Wf1T, H, NP, FFN_CH, CH, CH, FFN_CH, FFN_CH,
       (long long)NP * CH, 0, (long long)NP * FFN_CH,
       nullptr, 0, nullptr, 0, bf1, nullptr, 0, 0, 1);
  gemm(H, Wf2T, Z, NP, CH, FFN_CH, FFN_CH, CH, CH,
       (long long)NP * FFN_CH, 0, (long long)NP * CH,
       nullptr, 0, nullptr, 0, bf2, X1, (long long)NP * CH, CH, 0);

  // ---- final LN + head + outputs ----
  k_ln_final<<<BSZ * NP, 256, 0, stream>>>(Z, g_ln, b_ln, Wpred, bpred, ppos,
                                           out_x, out_np, out_center);
}